// Net_78907139162251
// MI455X (gfx1250) — compile-verified
//
#include <hip/hip_runtime.h>
#include <hip/hip_bf16.h>
#include <math.h>

// ---------------------------------------------------------------------------
// Mamba stack for MI455X (gfx1250).
// GEMMs use V_WMMA_F32_16X16X4_F32 (full fp32 matrix path, matches fp32
// reference closely). v4: LDS-staged, double-buffered GEMM with the tile fill
// done by GLOBAL_LOAD_ASYNC_TO_LDS_B128 (ASYNCcnt-tracked memory->LDS engine).
// The builtin takes int4* operands (learned from the round-3 diagnostic).
// ---------------------------------------------------------------------------

typedef __attribute__((ext_vector_type(2))) float v2f;
typedef __attribute__((ext_vector_type(8))) float v8f;
typedef __attribute__((ext_vector_type(4))) int   v4i;

#define AS1 __attribute__((address_space(1)))
#define AS3 __attribute__((address_space(3)))

#if defined(__has_builtin)
#if __has_builtin(__builtin_amdgcn_global_load_async_to_lds_b128)
#define HAVE_ASYNC_LDS 1
#endif
#endif
#ifndef HAVE_ASYNC_LDS
#define HAVE_ASYNC_LDS 0
#endif

#define BB   4
#define LL   1024
#define DIN  32
#define DD   512
#define NLAY 4
#define EDIM 1024
#define NST  16
#define KCV  4
#define RRK  32
#define TT   (BB * LL)   // 4096 tokens

#define KCH   32         // K-chunk staged in LDS per buffer
#define LDST  36         // LDS row stride (floats): conflict-free + 16B aligned

#if HAVE_ASYNC_LDS
__device__ __forceinline__ void async_copy_b128(const float* g, float* l)
{
    // (global src v4i*, LDS dst v4i*, imm offset, imm cpol)
    __builtin_amdgcn_global_load_async_to_lds_b128(
        (AS1 v4i*)(uintptr_t)g, (AS3 v4i*)(uintptr_t)l, 0, 0);
}
__device__ __forceinline__ void wait_async0()
{
#if __has_builtin(__builtin_amdgcn_s_wait_asynccnt)
    __builtin_amdgcn_s_wait_asynccnt(0);
#else
    asm volatile("s_wait_asynccnt 0x0" ::: "memory");
#endif
}
#endif

// ---------------------------------------------------------------------------
// fp32 WMMA GEMM:  C[TT x NF] = X[TT x KD] @ W[NF x KD]^T (+bias) (+Cin)
// Block = 128 threads = 4 waves. Block tile 64(M) x 64(N); each wave owns a
// 16(M) x 64(N) strip (4 accumulators). X/W tiles double-buffered in LDS.
// Fragment layouts per CDNA5 ISA:
//   A 16x4 f32 : lanes 0-15 = M rows; lane-half selects K pair (K0K1 | K2K3).
//   B 4x16 f32 : lane%16 = N col; lane-half selects K pair; VGPR = K lsb.
//   C/D 16x16  : VGPR r -> M = r + 8*(lane>=16), N = lane%16.
// ---------------------------------------------------------------------------
template <bool BIAS, bool ACC_IN>
__global__ __launch_bounds__(128) void wmma_gemm_f32(
    const float* __restrict__ X, int lda,
    const float* __restrict__ W, int ldw,
    const float* __restrict__ bias,
    float* __restrict__ C, int ldc,
    const float* __restrict__ Cin, int ldcin,
    int KD)
{
    __shared__ float lsX[2][64 * LDST];
    __shared__ float lsW[2][64 * LDST];

    const int tid    = threadIdx.x;
    const int lane   = tid & 31;
    const int wid    = tid >> 5;
    const int laneLo = lane & 15;
    const int laneHi = lane >> 4;
    const int tileM0 = blockIdx.x * 64;
    const int tileN0 = blockIdx.y * 64;

    // Cooperative-load assignment: 512 float4 segments per 64x32 tile,
    // 4 per thread. Thread covers rows tid/8 (+16,+32,+48), cols (tid%8)*4.
    const int r0 = tid >> 3;
    const int c0 = (tid & 7) * 4;

    v8f acc[4];
#pragma unroll
    for (int nt = 0; nt < 4; ++nt) {
#pragma unroll
        for (int r = 0; r < 8; ++r) {
            if (ACC_IN)
                acc[nt][r] = Cin[(size_t)(tileM0 + wid * 16 + r + 8 * laneHi) * ldcin +
                                 tileN0 + nt * 16 + laneLo];
            else
                acc[nt][r] = 0.0f;
        }
    }

    const int nch = KD / KCH;

#if HAVE_ASYNC_LDS
    // ---- async fill of chunk 0 ----
#pragma unroll
    for (int i = 0; i < 4; ++i) {
        const int row = r0 + i * 16;
        async_copy_b128(X + (size_t)(tileM0 + row) * lda + c0,
                        &lsX[0][row * LDST + c0]);
        async_copy_b128(W + (size_t)(tileN0 + row) * ldw + c0,
                        &lsW[0][row * LDST + c0]);
    }
    wait_async0();
    __syncthreads();

    for (int c = 0; c < nch; ++c) {
        const int cur = c & 1;
        // ---- async fill of chunk c+1 into the other buffer ----
        if (c + 1 < nch) {
            const int nxt = (c + 1) & 1;
            const int kbase = (c + 1) * KCH;
#pragma unroll
            for (int i = 0; i < 4; ++i) {
                const int row = r0 + i * 16;
                async_copy_b128(X + (size_t)(tileM0 + row) * lda + kbase + c0,
                                &lsX[nxt][row * LDST + c0]);
                async_copy_b128(W + (size_t)(tileN0 + row) * ldw + kbase + c0,
                                &lsW[nxt][row * LDST + c0]);
            }
        }
        // ---- compute chunk c from LDS (overlaps the async fill) ----
#pragma unroll
        for (int kk = 0; kk < KCH; kk += 4) {
            v2f a = *(const v2f*)&lsX[cur][(wid * 16 + laneLo) * LDST + kk + laneHi * 2];
#pragma unroll
            for (int nt = 0; nt < 4; ++nt) {
                v2f b = *(const v2f*)&lsW[cur][(nt * 16 + laneLo) * LDST + kk + laneHi * 2];
                acc[nt] = __builtin_amdgcn_wmma_f32_16x16x4_f32(
                    false, a, false, b, (short)0, acc[nt], false, false);
            }
        }
        if (c + 1 < nch) wait_async0();
        __syncthreads();
    }
#else
    // ---- fallback: register double-buffered staging ----
    float4 rx[4], rw[4];
#pragma unroll
    for (int i = 0; i < 4; ++i) {
        const int row = r0 + i * 16;
        rx[i] = *(const float4*)(X + (size_t)(tileM0 + row) * lda + c0);
        rw[i] = *(const float4*)(W + (size_t)(tileN0 + row) * ldw + c0);
    }
#pragma unroll
    for (int i = 0; i < 4; ++i) {
        const int row = r0 + i * 16;
        *(float4*)&lsX[0][row * LDST + c0] = rx[i];
        *(float4*)&lsW[0][row * LDST + c0] = rw[i];
    }
    __syncthreads();

    for (int c = 0; c < nch; ++c) {
        const int cur = c & 1;
        if (c + 1 < nch) {
            const int kbase = (c + 1) * KCH;
#pragma unroll
            for (int i = 0; i < 4; ++i) {
                const int row = r0 + i * 16;
                rx[i] = *(const float4*)(X + (size_t)(tileM0 + row) * lda + kbase + c0);
                rw[i] = *(const float4*)(W + (size_t)(tileN0 + row) * ldw + kbase + c0);
            }
        }
#pragma unroll
        for (int kk = 0; kk < KCH; kk += 4) {
            v2f a = *(const v2f*)&lsX[cur][(wid * 16 + laneLo) * LDST + kk + laneHi * 2];
#pragma unroll
            for (int nt = 0; nt < 4; ++nt) {
                v2f b = *(const v2f*)&lsW[cur][(nt * 16 + laneLo) * LDST + kk + laneHi * 2];
                acc[nt] = __builtin_amdgcn_wmma_f32_16x16x4_f32(
                    false, a, false, b, (short)0, acc[nt], false, false);
            }
        }
        if (c + 1 < nch) {
            const int nxt = (c + 1) & 1;
#pragma unroll
            for (int i = 0; i < 4; ++i) {
                const int row = r0 + i * 16;
                *(float4*)&lsX[nxt][row * LDST + c0] = rx[i];
                *(float4*)&lsW[nxt][row * LDST + c0] = rw[i];
            }
        }
        __syncthreads();
    }
#endif

#pragma unroll
    for (int nt = 0; nt < 4; ++nt) {
        const int col = tileN0 + nt * 16 + laneLo;
        float bv = BIAS ? bias[col] : 0.0f;
#pragma unroll
        for (int r = 0; r < 8; ++r) {
            const int row = tileM0 + wid * 16 + r + 8 * laneHi;
            C[(size_t)row * ldc + col] = acc[nt][r] + bv;
        }
    }
}

// ---------------------------------------------------------------------------
// RMSNorm: one 256-thread block per token (D=512).
// ---------------------------------------------------------------------------
__global__ __launch_bounds__(256) void rmsnorm_kernel(
    const float* __restrict__ h, const float* __restrict__ w,
    float* __restrict__ out)
{
    __shared__ float red[256];
    const int t = blockIdx.x;
    const float* row = h + (size_t)t * DD;
    float s = 0.0f;
    for (int d = threadIdx.x; d < DD; d += 256) {
        float v = row[d];
        s += v * v;
    }
    red[threadIdx.x] = s;
    __syncthreads();
    for (int off = 128; off > 0; off >>= 1) {
        if (threadIdx.x < off) red[threadIdx.x] += red[threadIdx.x + off];
        __syncthreads();
    }
    const float scale = rsqrtf(red[0] * (1.0f / DD) + 1e-5f);
    for (int d = threadIdx.x; d < DD; d += 256)
        out[(size_t)t * DD + d] = row[d] * scale * w[d];
}

// ---------------------------------------------------------------------------
// Causal depthwise conv (K=4) + bias + SiLU.  Reads xb half of xz (stride 2*ED).
// ---------------------------------------------------------------------------
__global__ __launch_bounds__(256) void conv_silu_kernel(
    const float* __restrict__ xz, const float* __restrict__ cw,
    const float* __restrict__ cb, float* __restrict__ xbc)
{
    const int idx = blockIdx.x * 256 + threadIdx.x;  // t*ED + e
    const int e = idx & (EDIM - 1);
    const int t = idx >> 10;
    const int p = t & (LL - 1);   // position within sequence
    const int base = t - p;       // first token of this batch's sequence
    float acc = cb[e];
#pragma unroll
    for (int j = 0; j < KCV; ++j) {
        const int tp = p - (KCV - 1) + j;
        if (tp >= 0)
            acc += xz[(size_t)(base + tp) * (2 * EDIM) + e] * cw[e * KCV + j];
    }
    xbc[idx] = acc / (1.0f + expf(-acc));  // silu
}

// ---------------------------------------------------------------------------
// Softplus in place over delta.
// ---------------------------------------------------------------------------
__global__ __launch_bounds__(256) void softplus_kernel(float* __restrict__ d)
{
    const int idx = blockIdx.x * 256 + threadIdx.x;
    const float v = d[idx];
    d[idx] = (v > 20.0f) ? v : log1pf(expf(v));
}

// ---------------------------------------------------------------------------
// Selective scan. Block = 256 threads = 16 channels x 16 states.
// Grid = (B, ED/16). Lane n holds state n; y reduced across 16 lanes.
// Writes y (+ x*D) into the xb half of xz (pre-conv data is dead by now).
// ---------------------------------------------------------------------------
__global__ __launch_bounds__(256) void scan_kernel(
    const float* __restrict__ xbc, const float* __restrict__ delta,
    const float* __restrict__ dbc, const float* __restrict__ A_log,
    const float* __restrict__ Dp, float* __restrict__ xz)
{
    const int n = threadIdx.x & 15;
    const int c = threadIdx.x >> 4;
    const int e = blockIdx.y * 16 + c;
    const int b = blockIdx.x;

    const float a = -expf(A_log[(size_t)e * NST + n]);
    const float dpar = Dp[e];
    float hst = 0.0f;

    for (int t = 0; t < LL; ++t) {
        const int tg = b * LL + t;
        const float dt = delta[(size_t)tg * EDIM + e];
        const float xv = xbc[(size_t)tg * EDIM + e];
        const float Bn = dbc[(size_t)tg * 64 + RRK + n];
        const float Cn = dbc[(size_t)tg * 64 + RRK + NST + n];
        const float dA = expf(dt * a);
        hst = dA * hst + (dt * xv) * Bn;
        float part = hst * Cn;
        part += __shfl_xor(part, 8, 16);
        part += __shfl_xor(part, 4, 16);
        part += __shfl_xor(part, 2, 16);
        part += __shfl_xor(part, 1, 16);
        if (n == 0)
            xz[(size_t)tg * (2 * EDIM) + e] = part + xv * dpar;
    }
}

// ---------------------------------------------------------------------------
// Gate: y *= silu(z), in place in the xb half of xz.
// ---------------------------------------------------------------------------
__global__ __launch_bounds__(256) void gate_kernel(float* __restrict__ xz)
{
    const int idx = blockIdx.x * 256 + threadIdx.x;  // t*ED + e
    const int e = idx & (EDIM - 1);
    const int t = idx >> 10;
    const size_t base = (size_t)t * (2 * EDIM);
    const float y = xz[base + e];
    const float z = xz[base + EDIM + e];
    xz[base + e] = y * (z / (1.0f + expf(-z)));
}

// ---------------------------------------------------------------------------
// fc2 + sigmoid: Dout = 1, one thread per token.
// ---------------------------------------------------------------------------
__global__ __launch_bounds__(256) void fc2_kernel(
    const float* __restrict__ h, const float* __restrict__ w,
    const float* __restrict__ b, float* __restrict__ out)
{
    const int t = blockIdx.x * 256 + threadIdx.x;
    if (t >= TT) return;
    float acc = b[0];
    for (int d = 0; d < DD; ++d) acc += h[(size_t)t * DD + d] * w[d];
    out[t] = 1.0f / (1.0f + expf(-acc));
}

// ---------------------------------------------------------------------------
extern "C" void kernel_launch(void* const* d_in, const int* in_sizes, int n_in,
                              void* d_out, int out_size, void* d_ws,
                              size_t ws_size, hipStream_t stream)
{
    (void)in_sizes; (void)n_in; (void)out_size; (void)ws_size;

    const float* x         = (const float*)d_in[0];   // (4,1024,32)
    const float* fc1_w     = (const float*)d_in[1];   // (512,32)
    const float* fc1_b     = (const float*)d_in[2];   // (512,)
    const float* fc2_w     = (const float*)d_in[3];   // (1,512)
    const float* fc2_b     = (const float*)d_in[4];   // (1,)
    const float* norm_w    = (const float*)d_in[5];   // (4,512)
    const float* in_proj_w = (const float*)d_in[6];   // (4,2048,512)
    const float* conv_w    = (const float*)d_in[7];   // (4,1024,4)
    const float* conv_b    = (const float*)d_in[8];   // (4,1024)
    const float* xproj_w   = (const float*)d_in[9];   // (4,64,1024)
    const float* dtproj_w  = (const float*)d_in[10];  // (4,1024,32)
    const float* dtproj_b  = (const float*)d_in[11];  // (4,1024)
    const float* A_log     = (const float*)d_in[12];  // (4,1024,16)
    const float* D_param   = (const float*)d_in[13];  // (4,1024)
    const float* out_proj  = (const float*)d_in[14];  // (4,512,1024)
    float* out = (float*)d_out;                       // (4096,)

    // Workspace carve (floats). Total ~21.2M floats = ~85 MB.
    float* ws    = (float*)d_ws;
    float* h     = ws;                          // TT*DD
    float* xn    = h + (size_t)TT * DD;         // TT*DD
    float* xz    = xn + (size_t)TT * DD;        // TT*2*ED
    float* xbc   = xz + (size_t)TT * 2 * EDIM;  // TT*ED
    float* dbc   = xbc + (size_t)TT * EDIM;     // TT*64
    float* delta = dbc + (size_t)TT * 64;       // TT*ED

    const dim3 blk(128);

    // fc1: h = x @ fc1_w^T + fc1_b    (KD=32, NF=512)
    wmma_gemm_f32<true, false><<<dim3(TT / 64, DD / 64), blk, 0, stream>>>(
        x, DIN, fc1_w, DIN, fc1_b, h, DD, nullptr, 0, DIN);

    for (int i = 0; i < NLAY; ++i) {
        const float* nw  = norm_w    + (size_t)i * DD;
        const float* ipw = in_proj_w + (size_t)i * 2 * EDIM * DD;
        const float* cw  = conv_w    + (size_t)i * EDIM * KCV;
        const float* cb  = conv_b    + (size_t)i * EDIM;
        const float* xpw = xproj_w   + (size_t)i * 64 * EDIM;
        const float* dtw = dtproj_w  + (size_t)i * EDIM * RRK;
        const float* dtb = dtproj_b  + (size_t)i * EDIM;
        const float* alg = A_log     + (size_t)i * EDIM * NST;
        const float* dpp = D_param   + (size_t)i * EDIM;
        const float* opw = out_proj  + (size_t)i * DD * EDIM;

        // xn = rmsnorm(h)
        rmsnorm_kernel<<<TT, 256, 0, stream>>>(h, nw, xn);

        // xz = xn @ in_proj_w^T   (KD=512, NF=2048)
        wmma_gemm_f32<false, false>
            <<<dim3(TT / 64, (2 * EDIM) / 64), blk, 0, stream>>>(
                xn, DD, ipw, DD, nullptr, xz, 2 * EDIM, nullptr, 0, DD);

        // xbc = silu(causal_dwconv(xb) + b)
        conv_silu_kernel<<<(TT * EDIM) / 256, 256, 0, stream>>>(xz, cw, cb, xbc);

        // dbc = xbc @ xproj_w^T   (KD=1024, NF=64)
        wmma_gemm_f32<false, false><<<dim3(TT / 64, 1), blk, 0, stream>>>(
            xbc, EDIM, xpw, EDIM, nullptr, dbc, 64, nullptr, 0, EDIM);

        // delta = dbc[:, :32] @ dtproj_w^T + dtproj_b   (KD=32, NF=1024)
        wmma_gemm_f32<true, false>
            <<<dim3(TT / 64, EDIM / 64), blk, 0, stream>>>(
                dbc, 64, dtw, RRK, dtb, delta, EDIM, nullptr, 0, RRK);

        // delta = softplus(delta)
        softplus_kernel<<<(TT * EDIM) / 256, 256, 0, stream>>>(delta);

        // selective scan -> y written into xb half of xz
        scan_kernel<<<dim3(BB, EDIM / 16), 256, 0, stream>>>(
            xbc, delta, dbc, alg, dpp, xz);

        // y *= silu(z) in place
        gate_kernel<<<(TT * EDIM) / 256, 256, 0, stream>>>(xz);

        // h += y @ out_proj_w^T   (KD=1024, NF=512, residual fused via ACC_IN)
        wmma_gemm_f32<false, true>
            <<<dim3(TT / 64, DD / 64), blk, 0, stream>>>(
                xz, 2 * EDIM, opw, EDIM, nullptr, h, DD, h, DD, EDIM);
    }

    // out = sigmoid(h @ fc2_w^T + fc2_b)
    fc2_kernel<<<TT / 256, 256, 0, stream>>>(h, fc2_w, fc2_b, out);
}